// TripletMarginLossOnlineMining_45475113730054
// MI455X (gfx1250) — compile-verified
//
#include <hip/hip_runtime.h>
#include <hip/hip_bf16.h>

// ---------------------------------------------------------------------------
// TripletMarginLoss with online (batch-all) mining, MI455X / gfx1250.
//
// Strategy:
//  * d^2(i,j) = |e_i|^2 + |e_j|^2 - 2*<e_i,e_j>.  The Gram matrix is computed
//    with the FP32 matrix pipe: v_wmma_f32_16x16x4_f32 (full fp32 precision,
//    no reason to drop precision on a latency-bound 256KB problem).
//  * Triplet sweep is restructured per-anchor: block b handles anchor i=b,
//    distance row + label flags live in LDS, deterministic tree reductions.
// ---------------------------------------------------------------------------

typedef __attribute__((ext_vector_type(2))) float v2f;
typedef __attribute__((ext_vector_type(8))) float v8f;

#define N 512
#define DIMS 128
#define MARGIN 1.0f
#define EPSF 1e-16f

// ---------------------------------------------------------------------------
// Kernel 0: squared L2 norms of each embedding row.
// ---------------------------------------------------------------------------
__global__ __launch_bounds__(256) void k_norms(const float* __restrict__ E,
                                               float* __restrict__ norms) {
    int row = blockIdx.x * blockDim.x + threadIdx.x;
    if (row >= N) return;
    const float4* p = (const float4*)(E + row * DIMS);
    float acc = 0.0f;
#pragma unroll
    for (int q = 0; q < DIMS / 4; ++q) {
        float4 v = p[q];
        acc += v.x * v.x + v.y * v.y + v.z * v.z + v.w * v.w;
    }
    norms[row] = acc;
}

// ---------------------------------------------------------------------------
// Kernel 1: distance matrix via FP32 WMMA.
// One wave per 16x16 tile of G = E * E^T.  Grid = (32, 32).
//
// V_WMMA_F32_16X16X4_F32 operand layout (ISA 7.12.2, 32-bit A 16x4):
//   lane L: M = L%16 ; VGPR0 holds K = 2*(L/16), VGPR1 holds K = 2*(L/16)+1.
// B (4x16) mirrors this transposed, so for E*E^T both operands are simply
// two consecutive floats from row (tileBase + L%16), column k0 + 2*(L/16).
// C/D layout: VGPR v, lane L -> M = v + 8*(L/16), Nc = L%16.
// ---------------------------------------------------------------------------
__global__ __launch_bounds__(32) void k_dist_wmma(const float* __restrict__ E,
                                                  const float* __restrict__ norms,
                                                  float* __restrict__ D) {
    const int bi = blockIdx.y;          // tile row  (anchor rows)
    const int bj = blockIdx.x;          // tile col
    const int lane = threadIdx.x;       // 0..31 (wave32)
    const int r    = lane & 15;
    const int kq   = (lane >> 4) << 1;  // 0 or 2

    const float* __restrict__ arow = E + (bi * 16 + r) * DIMS + kq;
    const float* __restrict__ brow = E + (bj * 16 + r) * DIMS + kq;

    v8f acc = {};
#pragma unroll
    for (int k0 = 0; k0 < DIMS; k0 += 4) {
        v2f a = *(const v2f*)(arow + k0);
        v2f b = *(const v2f*)(brow + k0);
        // 8 args: (neg_a, A, neg_b, B, c_mod, C, reuse_a, reuse_b)
        acc = __builtin_amdgcn_wmma_f32_16x16x4_f32(
            /*neg_a=*/false, a, /*neg_b=*/false, b,
            /*c_mod=*/(short)0, acc, /*reuse_a=*/false, /*reuse_b=*/false);
    }

    const int nc   = lane & 15;
    const int half = lane >> 4;
    const int gj   = bj * 16 + nc;
    const float nj = norms[gj];
#pragma unroll
    for (int v = 0; v < 8; ++v) {
        const int gi = bi * 16 + v + 8 * half;
        float sq = norms[gi] + nj - 2.0f * acc[v];
        float d  = (gi != gj && sq > 0.0f) ? __builtin_sqrtf(sq) : 0.0f;
        D[gi * N + gj] = d;
    }
}

// ---------------------------------------------------------------------------
// Kernel 2: per-anchor triplet sweep.  Block b == anchor i.
// For j in positives(i), k in negatives(i):  hinge(d_ij - d_ik + margin).
// n_triplets(i) = cp * cn computed analytically (j!=k is implied by labels).
// Deterministic: per-anchor partials, LDS tree reduction.
// ---------------------------------------------------------------------------
__global__ __launch_bounds__(256) void k_triplets(const float* __restrict__ D,
                                                  const int* __restrict__ labels,
                                                  float* __restrict__ p_sum,
                                                  float* __restrict__ p_pos,
                                                  float* __restrict__ p_trip) {
    __shared__ float drow[N];
    __shared__ int   same[N];
    __shared__ float s_sum[256];
    __shared__ float s_pos[256];
    __shared__ int   s_cp[256];
    __shared__ int   s_cn[256];

    const int i   = blockIdx.x;
    const int tid = threadIdx.x;
    const int li  = labels[i];

    for (int t = tid; t < N; t += 256) {
        drow[t] = D[i * N + t];
        same[t] = (labels[t] == li) ? 1 : 0;
    }
    __syncthreads();

    float sum = 0.0f, pos = 0.0f;
    int cp = 0, cn = 0;
    for (int j = tid; j < N; j += 256) {
        if (j == i) continue;
        if (same[j]) {
            ++cp;
            const float dij = drow[j] + MARGIN;
            for (int k = 0; k < N; ++k) {
                if (!same[k]) {
                    const float val = dij - drow[k];
                    if (val > EPSF) { sum += val; pos += 1.0f; }
                    else if (val > 0.0f) { sum += val; }
                }
            }
        } else {
            ++cn;   // negatives of anchor i (i itself excluded: same label)
        }
    }

    s_sum[tid] = sum; s_pos[tid] = pos; s_cp[tid] = cp; s_cn[tid] = cn;
    __syncthreads();
#pragma unroll
    for (int off = 128; off > 0; off >>= 1) {
        if (tid < off) {
            s_sum[tid] += s_sum[tid + off];
            s_pos[tid] += s_pos[tid + off];
            s_cp[tid]  += s_cp[tid + off];
            s_cn[tid]  += s_cn[tid + off];
        }
        __syncthreads();
    }
    if (tid == 0) {
        p_sum[i]  = s_sum[0];
        p_pos[i]  = s_pos[0];
        p_trip[i] = (float)(s_cp[0] * s_cn[0]);
    }
}

// ---------------------------------------------------------------------------
// Kernel 3: final 512 -> 2 reduction.  out[0]=loss, out[1]=fraction_positive.
// ---------------------------------------------------------------------------
__global__ __launch_bounds__(256) void k_finalize(const float* __restrict__ p_sum,
                                                  const float* __restrict__ p_pos,
                                                  const float* __restrict__ p_trip,
                                                  float* __restrict__ out) {
    __shared__ float s_sum[256];
    __shared__ float s_pos[256];
    __shared__ float s_trip[256];
    const int tid = threadIdx.x;
    s_sum[tid]  = p_sum[tid]  + p_sum[tid + 256];
    s_pos[tid]  = p_pos[tid]  + p_pos[tid + 256];
    s_trip[tid] = p_trip[tid] + p_trip[tid + 256];
    __syncthreads();
#pragma unroll
    for (int off = 128; off > 0; off >>= 1) {
        if (tid < off) {
            s_sum[tid]  += s_sum[tid + off];
            s_pos[tid]  += s_pos[tid + off];
            s_trip[tid] += s_trip[tid + off];
        }
        __syncthreads();
    }
    if (tid == 0) {
        out[0] = s_sum[0] / (s_pos[0] + EPSF);   // loss
        out[1] = s_pos[0] / (s_trip[0] + EPSF);  // fraction_positive
    }
}

// ---------------------------------------------------------------------------
// Launch
// ---------------------------------------------------------------------------
extern "C" void kernel_launch(void* const* d_in, const int* in_sizes, int n_in,
                              void* d_out, int out_size, void* d_ws, size_t ws_size,
                              hipStream_t stream) {
    const float* E      = (const float*)d_in[0];   // [512,128] fp32
    const int*   labels = (const int*)d_in[1];     // [512] int32
    float*       out    = (float*)d_out;           // [2] fp32

    float* ws     = (float*)d_ws;
    float* norms  = ws;                 // 512
    float* D      = ws + N;             // 512*512
    float* p_sum  = D + N * N;          // 512
    float* p_pos  = p_sum + N;          // 512
    float* p_trip = p_pos + N;          // 512

    k_norms<<<dim3(2), dim3(256), 0, stream>>>(E, norms);
    k_dist_wmma<<<dim3(N / 16, N / 16), dim3(32), 0, stream>>>(E, norms, D);
    k_triplets<<<dim3(N), dim3(256), 0, stream>>>(D, labels, p_sum, p_pos, p_trip);
    k_finalize<<<dim3(1), dim3(256), 0, stream>>>(p_sum, p_pos, p_trip, out);
}